// SC_Loss_61478161875588
// MI455X (gfx1250) — compile-verified
//
#include <hip/hip_runtime.h>
#include <stdint.h>

// Problem constants from the reference (x: [B, C], C fixed at 4096).
#define CDIM 4096
#define TPB  256

// Types matching the builtin's signature: vV4i*1 / vV4i*3 (gcc vector_size).
typedef int v4i_vs __attribute__((vector_size(16)));
typedef __attribute__((address_space(1))) v4i_vs* g_v4i_ptr;
typedef __attribute__((address_space(3))) v4i_vs* l_v4i_ptr;

// gfx1250 async global->LDS copy of 16 bytes (GLOBAL_LOAD_ASYNC_TO_LDS_B128,
// tracked by ASYNCcnt). Builtin if the toolchain declares it, else inline asm.
__device__ __forceinline__ void async_g2l_b128(const void* gsrc, void* ldst) {
#if __has_builtin(__builtin_amdgcn_global_load_async_to_lds_b128)
  __builtin_amdgcn_global_load_async_to_lds_b128(
      (g_v4i_ptr)gsrc, (l_v4i_ptr)ldst, /*imm offset*/0, /*cpol*/0);
#else
  asm volatile("global_load_async_to_lds_b128 %0, %1, off"
               :
               : "v"((unsigned)(uintptr_t)ldst),
                 "v"((unsigned long long)(uintptr_t)gsrc)
               : "memory");
#endif
}

__device__ __forceinline__ void wait_async0() {
#if __has_builtin(__builtin_amdgcn_s_wait_asynccnt)
  __builtin_amdgcn_s_wait_asynccnt(0);
#else
  asm volatile("s_wait_asynccnt 0" ::: "memory");
#endif
}

// One workgroup per batch row b:
//  - async-stage x[b,:] (16KB) and distance_rank[target[b],:] (16KB) into LDS
//  - CE: block max + sum-exp over the LDS row
//  - SC: 4093 LDS gathers + log1p/logaddexp, block-summed
//  - write one combined partial per row (deterministic: no FP atomics)
__global__ __launch_bounds__(TPB) void sc_loss_row_kernel(
    const float* __restrict__ x,
    const int*   __restrict__ target,
    const int*   __restrict__ drank,
    const int*   __restrict__ kidx,
    float*       __restrict__ partial,
    int B) {
  __shared__ float sx[CDIM];   // 16 KB: x row
  __shared__ int   sr[CDIM];   // 16 KB: ranks row
  __shared__ float sred[TPB];  // 1 KB : reductions

  const int b   = blockIdx.x;
  const int tid = threadIdx.x;
  const int t   = target[b];

  const float* xrow = x     + (size_t)b * CDIM;
  const int*   rrow = drank + (size_t)t * CDIM;
  const int*   krow = kidx  + (size_t)b * (CDIM - 3);

  // Stage both 16KB rows into LDS: 256 lanes x 4 iters x 16B per row.
  for (int i = tid; i < CDIM / 4; i += TPB) {
    async_g2l_b128(xrow + i * 4, &sx[i * 4]);
    async_g2l_b128(rrow + i * 4, &sr[i * 4]);
  }
  // While the async DMA drains, warm L2/L0 for the coalesced k_idx stream
  // (4093 ints ~= 16KB ~= 128 cachelines).
  if (tid < 128) __builtin_prefetch(krow + tid * 32, 0, 3);

  wait_async0();       // this wave's ASYNCcnt -> 0
  __syncthreads();     // all waves' LDS writes visible

  // ---------------- cross-entropy term ----------------
  float m = -INFINITY;
  for (int i = tid; i < CDIM; i += TPB) m = fmaxf(m, sx[i]);
  sred[tid] = m;
  __syncthreads();
  for (int off = TPB / 2; off > 0; off >>= 1) {
    if (tid < off) sred[tid] = fmaxf(sred[tid], sred[tid + off]);
    __syncthreads();
  }
  m = sred[0];
  __syncthreads();

  float se = 0.0f;
  for (int i = tid; i < CDIM; i += TPB) se += expf(sx[i] - m);
  sred[tid] = se;
  __syncthreads();
  for (int off = TPB / 2; off > 0; off >>= 1) {
    if (tid < off) sred[tid] += sred[tid + off];
    __syncthreads();
  }
  const float lse  = logf(sred[0]);
  const float ce_b = -(sx[t] - m - lse);  // -log_softmax(x)[b, target[b]]
  __syncthreads();                        // before reusing sred

  // ---------------- SC term ----------------
  const float alpha     = (float)(CDIM - 1);
  const float inv_alpha = 1.0f / alpha;
  const float e         = -0.5f * (alpha + 1.0f);  // -2048
  const float yi        = sx[sr[1]];

  float ss = 0.0f;  // sum of (la - logaddexp(la, lb)) over this thread's j's
  for (int j = tid; j < CDIM - 3; j += TPB) {
    const float yj = sx[sr[j + 2]];
    const int   k  = krow[j];          // in [j+3, C-1], valid index
    const float yk = sx[sr[k]];
    const float dj = yi - yj;
    const float dk = yi - yk;
    const float la = e * log1pf(dj * dj * inv_alpha);
    const float lb = e * log1pf(dk * dk * inv_alpha);
    const float hi = fmaxf(la, lb);
    const float lo = fminf(la, lb);
    const float lae = hi + log1pf(expf(lo - hi));  // logaddexp(la, lb)
    ss += (la - lae);
  }
  sred[tid] = ss;
  __syncthreads();
  for (int off = TPB / 2; off > 0; off >>= 1) {
    if (tid < off) sred[tid] += sred[tid + off];
    __syncthreads();
  }

  if (tid == 0) {
    const float inv_ln2  = 1.44269504088896340736f;
    const float sum_log2 = sred[0] * inv_ln2;   // sum of log2(pijk) for row b
    const float theta    = 0.6f;
    const float Bf       = (float)B;
    // theta*ce/B  +  (1-theta) * (-sum_log2) / (B*(C-1))
    partial[b] = theta * ce_b / Bf
               + (1.0f - theta) * (-sum_log2) / (Bf * alpha);
  }
}

__global__ __launch_bounds__(TPB) void sc_loss_reduce_kernel(
    const float* __restrict__ partial, float* __restrict__ out, int n) {
  __shared__ float sred[TPB];
  float s = 0.0f;
  for (int i = (int)threadIdx.x; i < n; i += TPB) s += partial[i];
  sred[threadIdx.x] = s;
  __syncthreads();
  for (int off = TPB / 2; off > 0; off >>= 1) {
    if ((int)threadIdx.x < off) sred[threadIdx.x] += sred[threadIdx.x + off];
    __syncthreads();
  }
  if (threadIdx.x == 0) out[0] = sred[0];
}

extern "C" void kernel_launch(void* const* d_in, const int* in_sizes, int n_in,
                              void* d_out, int out_size, void* d_ws, size_t ws_size,
                              hipStream_t stream) {
  const float* x      = (const float*)d_in[0];  // [B, C] f32
  const int*   target = (const int*)  d_in[1];  // [B] i32
  const int*   drank  = (const int*)  d_in[2];  // [C, C] i32
  const int*   kidx   = (const int*)  d_in[3];  // [B, C-3] i32
  const int B = in_sizes[1];                    // 8192

  float* partial = (float*)d_ws;                // B floats (32 KB) of scratch

  sc_loss_row_kernel<<<B, TPB, 0, stream>>>(x, target, drank, kidx, partial, B);
  sc_loss_reduce_kernel<<<1, TPB, 0, stream>>>(partial, (float*)d_out, B);
}